// CausalSelfAttention_25254407700588
// MI455X (gfx1250) — compile-verified
//
#include <hip/hip_runtime.h>
#include <math.h>

// Problem constants (B, T, D, H) = (4, 2048, 1024, 16)
#define Bb    4
#define Tt    2048
#define Dd    1024
#define Hh    16
#define DHd   64
#define HALFd 32

typedef __attribute__((ext_vector_type(2))) float v2f;
typedef __attribute__((ext_vector_type(8))) float v8f;

// CDNA5 fp32 WMMA: D(16x16,f32) = A(16x4,f32) * B(4x16,f32) + C
// 8-arg form: (neg_a, A, neg_b, B, c_mod, C, reuse_a, reuse_b)
__device__ __forceinline__ v8f wmma_f32(v2f a, v2f b, v8f c) {
  return __builtin_amdgcn_wmma_f32_16x16x4_f32(false, a, false, b, (short)0, c,
                                               false, false);
}

// ds_swizzle xor-butterfly within 16-lane row groups (wave32)
#define SWZ(x, imm) \
  __int_as_float(__builtin_amdgcn_ds_swizzle(__float_as_int(x), (imm)))

__device__ __forceinline__ float rowmax16(float v) {
  v = fmaxf(v, SWZ(v, 0x041F));  // xor 1
  v = fmaxf(v, SWZ(v, 0x081F));  // xor 2
  v = fmaxf(v, SWZ(v, 0x101F));  // xor 4
  v = fmaxf(v, SWZ(v, 0x201F));  // xor 8
  return v;
}
__device__ __forceinline__ float rowsum16(float v) {
  v += SWZ(v, 0x041F);
  v += SWZ(v, 0x081F);
  v += SWZ(v, 0x101F);
  v += SWZ(v, 0x201F);
  return v;
}

// --------------------------------------------------------------------------
// Kernel 1: qkv = x @ W_qkv^T, split into q/k/v (B,H,T,DH), RoPE on q,k,
//           fold DH^-0.5 into q. One wave per 16(t) x 64(d) tile (one head).
// --------------------------------------------------------------------------
__global__ __launch_bounds__(32)
void qkv_rope_kernel(const float* __restrict__ x, const float* __restrict__ cosT,
                     const float* __restrict__ sinT, const float* __restrict__ Wqkv,
                     float* __restrict__ qw, float* __restrict__ kw,
                     float* __restrict__ vw) {
  const int lane  = threadIdx.x;
  const int m0    = blockIdx.x * 16;   // row tile over B*T
  const int n0    = blockIdx.y * 64;   // col chunk over 3*D (one head of q/k/v)
  const int nrow  = lane & 15;
  const int hi16  = lane >> 4;
  const int khalf = hi16 * 2;

  v8f acc[4] = {};
  const float* arow = x    + (size_t)(m0 + nrow) * Dd;
  const float* brow = Wqkv + (size_t)(n0 + nrow) * Dd;
#pragma unroll 4
  for (int k0 = 0; k0 < Dd; k0 += 4) {
    const int kk = k0 + khalf;
    v2f a = *(const v2f*)(arow + kk);           // A[m][k..k+1]
#pragma unroll
    for (int j = 0; j < 4; ++j) {
      v2f b = *(const v2f*)(brow + (size_t)j * 16 * Dd + kk);  // B[k][n]=W[n][k]
      acc[j] = wmma_f32(a, b, acc[j]);
    }
  }

  const int sec = n0 / Dd;             // 0=q, 1=k, 2=v
  const int h   = (n0 % Dd) / DHd;
  const int b   = m0 / Tt;
  const int t0  = m0 % Tt;
  float* dst = (sec == 0 ? qw : (sec == 1 ? kw : vw)) +
               ((size_t)(b * Hh + h) * Tt + t0) * DHd;

  if (sec < 2) {  // RoPE: pairs (acc[p], acc[p+2]) share angle index p*16+nrow
#pragma unroll
    for (int r = 0; r < 8; ++r) {
      const int M = r + (hi16 << 3);
      const int t = t0 + M;
#pragma unroll
      for (int p = 0; p < 2; ++p) {
        const int idx = p * 16 + nrow;
        const float c = cosT[t * HALFd + idx];
        const float s = sinT[t * HALFd + idx];
        const float lo = acc[p][r], hi = acc[p + 2][r];
        acc[p][r]     = lo * c - hi * s;
        acc[p + 2][r] = lo * s + hi * c;
      }
    }
  }
  const float qs = (sec == 0) ? 0.125f : 1.0f;   // fold DH^-0.5 into q
#pragma unroll
  for (int r = 0; r < 8; ++r) {
    const int M = r + (hi16 << 3);
#pragma unroll
    for (int j = 0; j < 4; ++j)
      dst[(size_t)M * DHd + j * 16 + nrow] = acc[j][r] * qs;
  }
}

// --------------------------------------------------------------------------
// Kernel 2: flash-style causal attention, one wave per (b,h,16-row q tile).
//           out = softmax(qk^T) @ v - diag(w)*v, written to (B,H,T,DH).
// --------------------------------------------------------------------------
__global__ __launch_bounds__(32)
void attn_kernel(const float* __restrict__ qw, const float* __restrict__ kw,
                 const float* __restrict__ vw, float* __restrict__ aw) {
  __shared__ float pmat[16 * 16];
  __shared__ float sdiag[16];

  const int lane  = threadIdx.x;
  const int bh    = blockIdx.x;        // b*H + h
  const int qt    = blockIdx.y;        // q tile index
  const int t0    = qt * 16;
  const int nrow  = lane & 15;
  const int hi16  = lane >> 4;
  const int khalf = hi16 * 2;

  const float* qbase = qw + (size_t)bh * Tt * DHd;
  const float* kbase = kw + (size_t)bh * Tt * DHd;
  const float* vbase = vw + (size_t)bh * Tt * DHd;

  // Preload Q A-fragments for all 16 K-slices over DH=64 (loop-invariant)
  v2f qa[16];
  {
    const float* qrow = qbase + (size_t)(t0 + nrow) * DHd;
#pragma unroll
    for (int s = 0; s < 16; ++s) qa[s] = *(const v2f*)(qrow + s * 4 + khalf);
  }

  v8f O[4] = {};
  float mrow[8], lrow[8];
#pragma unroll
  for (int r = 0; r < 8; ++r) { mrow[r] = -__builtin_inff(); lrow[r] = 0.f; }

  for (int jt = 0; jt <= qt; ++jt) {
    const int tk0 = jt * 16;
    // S = Q @ K^T  (scale already folded into q)
    v8f S = {};
    const float* krow = kbase + (size_t)(tk0 + nrow) * DHd;
#pragma unroll
    for (int s = 0; s < 16; ++s) {
      v2f bf = *(const v2f*)(krow + s * 4 + khalf);  // B[d][n]=k[n][d]
      S = wmma_f32(qa[s], bf, S);
    }

    const bool diag = (jt == qt);
    float alpha[8];
#pragma unroll
    for (int r = 0; r < 8; ++r) {
      const int M = r + (hi16 << 3);
      float sv = S[r];
      if (diag && nrow > M) sv = -__builtin_inff();   // causal mask
      const float rm   = rowmax16(sv);
      const float mnew = fmaxf(mrow[r], rm);
      const float a    = __expf(mrow[r] - mnew);
      const float p    = __expf(sv - mnew);
      lrow[r] = lrow[r] * a + rowsum16(p);
      mrow[r] = mnew;
      alpha[r] = a;
      pmat[M * 16 + nrow] = p;
      if (diag && nrow == M) sdiag[M] = p;            // w_ii (unnormalized)
    }
    __syncthreads();

#pragma unroll
    for (int j = 0; j < 4; ++j)
#pragma unroll
      for (int r = 0; r < 8; ++r) O[j][r] *= alpha[r];

    // O += P @ V  (P re-fragmented through LDS into A layout)
#pragma unroll
    for (int s4 = 0; s4 < 4; ++s4) {
      v2f pa = *(const v2f*)(&pmat[nrow * 16 + s4 * 4 + khalf]);
      const float* vrow = vbase + (size_t)(tk0 + s4 * 4 + khalf) * DHd;
#pragma unroll
      for (int j = 0; j < 4; ++j) {
        v2f bf = { vrow[j * 16 + nrow], vrow[DHd + j * 16 + nrow] };
        O[j] = wmma_f32(pa, bf, O[j]);
      }
    }
    __syncthreads();
  }

  float* dst = aw + (size_t)bh * Tt * DHd;
#pragma unroll
  for (int r = 0; r < 8; ++r) {
    const int M = r + (hi16 << 3);
    const float inv = 1.0f / lrow[r];
    const float pd  = sdiag[M];
#pragma unroll
    for (int j = 0; j < 4; ++j) {
      const int d = j * 16 + nrow;
      const float vs = vbase[(size_t)(t0 + M) * DHd + d];
      dst[(size_t)(t0 + M) * DHd + d] = (O[j][r] - pd * vs) * inv;
    }
  }
}

// --------------------------------------------------------------------------
// Kernel 3: out = attn @ W_proj^T ; attn read from head-interleaved layout.
// --------------------------------------------------------------------------
__global__ __launch_bounds__(32)
void proj_kernel(const float* __restrict__ aw, const float* __restrict__ Wp,
                 float* __restrict__ out) {
  const int lane  = threadIdx.x;
  const int m0    = blockIdx.x * 16;
  const int n0    = blockIdx.y * 64;
  const int nrow  = lane & 15;
  const int hi16  = lane >> 4;
  const int khalf = hi16 * 2;
  const int b     = m0 / Tt;
  const int t     = (m0 % Tt) + nrow;

  v8f acc[4] = {};
  const float* abase = aw + ((size_t)b * Hh * Tt + t) * DHd;  // + h*Tt*DHd + d
  const float* brow  = Wp + (size_t)(n0 + nrow) * Dd;
#pragma unroll 2
  for (int k0 = 0; k0 < Dd; k0 += 4) {
    const int kk = k0 + khalf;
    const int h = kk >> 6, d = kk & 63;     // kk,kk+1 stay within one head
    v2f a = *(const v2f*)(abase + (size_t)h * Tt * DHd + d);
#pragma unroll
    for (int j = 0; j < 4; ++j) {
      v2f bb = *(const v2f*)(brow + (size_t)j * 16 * Dd + kk);
      acc[j] = wmma_f32(a, bb, acc[j]);
    }
  }
  float* orow = out + (size_t)m0 * Dd + n0;
#pragma unroll
  for (int r = 0; r < 8; ++r) {
    const int M = r + (hi16 << 3);
#pragma unroll
    for (int j = 0; j < 4; ++j)
      orow[(size_t)M * Dd + j * 16 + nrow] = acc[j][r];
  }
}

// --------------------------------------------------------------------------
extern "C" void kernel_launch(void* const* d_in, const int* in_sizes, int n_in,
                              void* d_out, int out_size, void* d_ws,
                              size_t ws_size, hipStream_t stream) {
  (void)in_sizes; (void)n_in; (void)out_size; (void)ws_size;
  const float* x     = (const float*)d_in[0];
  const float* cosT  = (const float*)d_in[1];
  const float* sinT  = (const float*)d_in[2];
  const float* Wqkv  = (const float*)d_in[3];
  const float* Wproj = (const float*)d_in[4];
  float* out = (float*)d_out;
  float* ws  = (float*)d_ws;

  const size_t per = (size_t)Bb * Hh * Tt * DHd;  // 8,388,608 floats
  float* qw = ws;                                  // ws usage: 4*per*4B = 128MB
  float* kw = ws + per;
  float* vw = ws + 2 * per;
  float* aw = ws + 3 * per;

  dim3 g1(Bb * Tt / 16, (3 * Dd) / 64);  // 512 x 48
  qkv_rope_kernel<<<g1, 32, 0, stream>>>(x, cosT, sinT, Wqkv, qw, kw, vw);

  dim3 g2(Bb * Hh, Tt / 16);             // 64 x 128
  attn_kernel<<<g2, 32, 0, stream>>>(qw, kw, vw, aw);

  dim3 g3(Bb * Tt / 16, Dd / 64);        // 512 x 16
  proj_kernel<<<g3, 32, 0, stream>>>(aw, Wproj, out);
}